// OFTRotationModule_75333726372428
// MI455X (gfx1250) — compile-verified
//
#include <hip/hip_runtime.h>

typedef float v2f __attribute__((ext_vector_type(2)));
typedef float v8f __attribute__((ext_vector_type(8)));

#define NVEC 2016  // 64*63/2 packed strict-upper-triangle elements per block

// ---------------------------------------------------------------------------
// Phase 1: build the 64 rotation matrices R = I + 2Q + 2Q^2 + 2Q^3 + Q^4
// and store them TRANSPOSED: Rt[r][c*64 + k] = R[r][k][c]  (1 MB in d_ws).
// One workgroup per rotation block; all matmuls are 64x64 in LDS (trivial cost).
// ---------------------------------------------------------------------------
__global__ __launch_bounds__(256) void oft_build_R(const float* __restrict__ w,
                                                   float* __restrict__ Rt) {
  const int r = blockIdx.x;   // 0..63
  const int t = threadIdx.x;  // 0..255

  __shared__ float sQ[64 * 65];   // padded to kill bank conflicts
  __shared__ float sP[64 * 65];   // current power Q^p
  __shared__ float sPn[64 * 65];  // next power

  const float* wr = w + r * NVEC;

  int ii[16], jj[16];
  float racc[16];

  // Build skew-symmetric Q from packed upper triangle; init R = I + 2Q.
  #pragma unroll
  for (int s = 0; s < 16; ++s) {
    const int idx = t + 256 * s;
    const int i = idx >> 6, j = idx & 63;
    ii[s] = i; jj[s] = j;
    float v;
    if (i < j)      v =  wr[i * 63 - (i * (i - 1)) / 2 + (j - i - 1)];
    else if (i > j) v = -wr[j * 63 - (j * (j - 1)) / 2 + (i - j - 1)];
    else            v = 0.0f;
    sQ[i * 65 + j] = v;
    sP[i * 65 + j] = v;
    racc[s] = (i == j ? 1.0f : 0.0f) + 2.0f * v;
  }
  __syncthreads();

  // R += 2*Q^2 + 2*Q^3 + 1*Q^4
  for (int p = 2; p <= 4; ++p) {
    #pragma unroll
    for (int s = 0; s < 16; ++s) {
      const int i = ii[s], j = jj[s];
      float acc = 0.0f;
      #pragma unroll 8
      for (int k = 0; k < 64; ++k) acc += sP[i * 65 + k] * sQ[k * 65 + j];
      sPn[i * 65 + j] = acc;
    }
    __syncthreads();
    const float alpha = (p == 4) ? 1.0f : 2.0f;
    #pragma unroll
    for (int s = 0; s < 16; ++s) {
      const int i = ii[s], j = jj[s];
      const float v = sPn[i * 65 + j];
      racc[s] += alpha * v;
      sP[i * 65 + j] = v;
    }
    __syncthreads();
  }

  // Store transposed so B-fragments read consecutive K from LDS later.
  float* outr = Rt + r * 4096;
  #pragma unroll
  for (int s = 0; s < 16; ++s) outr[jj[s] * 64 + ii[s]] = racc[s];
}

// ---------------------------------------------------------------------------
// Phase 2: out[row][r*64 + c] = sum_k x[row][r*64 + k] * R[r][k][c]
// Grid: (64 r-blocks, 128 row-groups of 128 rows). 256 threads = 8 waves.
// Each wave computes a 16x64 output tile via 4 x (16 chained
// v_wmma_f32_16x16x4_f32) with K = 64. Full fp32, HBM-bandwidth bound.
// ---------------------------------------------------------------------------
__global__ __launch_bounds__(256) void oft_apply(const float* __restrict__ x,
                                                 const float* __restrict__ Rt,
                                                 float* __restrict__ out) {
  const int rb   = blockIdx.x;     // rotation block 0..63
  const int mg   = blockIdx.y;     // row group 0..127
  const int t    = threadIdx.x;
  const int lane = t & 31;
  const int wv   = t >> 5;         // wave id 0..7
  const int half = lane >> 4;      // 0 for lanes 0-15, 1 for lanes 16-31
  const int l15  = lane & 15;

  __shared__ float sB[64 * 66];       // Rt[c][k], stride 66 (8B aligned rows)
  __shared__ float sA[8][16 * 68];    // per-wave x tile [m][k], stride 68 (16B aligned)

  // Stage B = Rt[rb] (coalesced global, conflict-free LDS writes).
  const float* Rtr = Rt + rb * 4096;
  #pragma unroll
  for (int s = 0; s < 16; ++s) {
    const int idx = t + 256 * s;      // idx = c*64 + k
    const int c = idx >> 6, k = idx & 63;
    sB[c * 66 + k] = Rtr[idx];
  }

  // Stage A: each wave loads its own 16x64 tile with float4 (256B/row segments).
  const int row0 = mg * 128 + wv * 16;
  #pragma unroll
  for (int i = 0; i < 8; ++i) {
    const int flat = i * 128 + lane * 4;      // 128 floats per pass
    const int rr = flat >> 6, ck = flat & 63;
    const float4 v =
        *(const float4*)(x + (size_t)(row0 + rr) * 4096 + rb * 64 + ck);
    *(float4*)&sA[wv][rr * 68 + ck] = v;
  }
  __syncthreads();

  // 4 N-tiles of 16 columns each; K = 64 -> 16 chained WMMAs per tile.
  #pragma unroll
  for (int n = 0; n < 4; ++n) {
    v8f acc = {};
    #pragma unroll
    for (int kk = 0; kk < 16; ++kk) {
      // A frag: lane l15 holds row M=l15, K = 4*kk + 2*half + {0,1}
      v2f a = *(const v2f*)&sA[wv][l15 * 68 + kk * 4 + 2 * half];
      // B frag: lane l15 holds col N=n*16+l15, K = 4*kk + 2*half + {0,1}
      v2f b = *(const v2f*)&sB[(n * 16 + l15) * 66 + kk * 4 + 2 * half];
      acc = __builtin_amdgcn_wmma_f32_16x16x4_f32(false, a, false, b,
                                                  (short)0, acc, false, false);
    }
    // D layout: VGPR v = row (v + 8*half), col = n*16 + l15.
    float* o = out + (size_t)(row0 + 8 * half) * 4096 + rb * 64 + n * 16 + l15;
    #pragma unroll
    for (int v = 0; v < 8; ++v) o[(size_t)v * 4096] = acc[v];
  }
}

extern "C" void kernel_launch(void* const* d_in, const int* in_sizes, int n_in,
                              void* d_out, int out_size, void* d_ws, size_t ws_size,
                              hipStream_t stream) {
  (void)in_sizes; (void)n_in; (void)out_size; (void)ws_size;
  const float* x = (const float*)d_in[0];   // [4,4096,4096] f32
  const float* w = (const float*)d_in[1];   // [64,2016]     f32
  float* out = (float*)d_out;               // [4,4096,4096] f32
  float* Rt  = (float*)d_ws;                // 64*64*64 f32 = 1 MB scratch

  oft_build_R<<<dim3(64), dim3(256), 0, stream>>>(w, Rt);
  oft_apply<<<dim3(64, 128), dim3(256), 0, stream>>>(x, Rt, out);
}